// GCN_764504178704
// MI455X (gfx1250) — compile-verified
//
#include <hip/hip_runtime.h>
#include <hip/hip_bf16.h>

#define DIM 128

typedef float v2f __attribute__((ext_vector_type(2)));
typedef float v8f __attribute__((ext_vector_type(8)));

// ---------------------------------------------------------------------------
// Hardware tanh: CDNA5 V_TANH_F32 (single trans-unit op).
// ---------------------------------------------------------------------------
__device__ __forceinline__ float fast_tanh(float v)
{
#if __has_builtin(__builtin_amdgcn_tanhf)
    return __builtin_amdgcn_tanhf(v);
#elif __has_builtin(__builtin_amdgcn_tanh_f32)
    return __builtin_amdgcn_tanh_f32(v);
#else
    float r;
    asm volatile("v_tanh_f32 %0, %1" : "=v"(r) : "v"(v));
    return r;
#endif
}

// ---------------------------------------------------------------------------
// Phase 1: h = tanh(x @ W), f32 WMMA 16x16x4.
// Block = 256 threads = 8 waves; each wave computes a 16-row x 128-col strip.
// W (128x128 f32 = 64 KB) staged in LDS once per block.
// ---------------------------------------------------------------------------
__global__ __launch_bounds__(256) void gcn_gemm_tanh_wmma(
    const float* __restrict__ x, const float* __restrict__ W,
    float* __restrict__ h, int n_nodes)
{
    __shared__ float Ws[DIM * DIM];   // 64 KB of the 320 KB WGP LDS

    // Cooperative coalesced load of W into LDS (float4).
    for (int i = threadIdx.x; i < DIM * DIM / 4; i += 256) {
        ((float4*)Ws)[i] = ((const float4*)W)[i];
    }
    __syncthreads();

    const int lane = threadIdx.x & 31;
    const int wave = threadIdx.x >> 5;
    const int half = lane >> 4;        // 0: lanes 0-15, 1: lanes 16-31
    const int l16  = lane & 15;

    const int row_base = blockIdx.x * 128 + wave * 16;

    // A-matrix row this lane feeds (clamped: WMMA needs EXEC all-ones).
    int arow = row_base + l16;
    if (arow >= n_nodes) arow = n_nodes - 1;
    const float* xrow = x + (size_t)arow * DIM;

    v8f zero = {};
    v8f acc[8];
#pragma unroll
    for (int t = 0; t < 8; ++t) acc[t] = zero;

    for (int k0 = 0; k0 < DIM; k0 += 4) {
        // A 16x4 tile: lane holds A[m=l16][k0 + 2*half + {0,1}]
        const int kl = k0 + 2 * half;
        v2f a;
        a.x = xrow[kl + 0];
        a.y = xrow[kl + 1];
#pragma unroll
        for (int t = 0; t < 8; ++t) {
            // B 4x16 tile of W: lane holds W[k0 + 2*half + {0,1}][n = t*16 + l16]
            const int n = t * 16 + l16;
            v2f bm;
            bm.x = Ws[(kl + 0) * DIM + n];
            bm.y = Ws[(kl + 1) * DIM + n];
            acc[t] = __builtin_amdgcn_wmma_f32_16x16x4_f32(
                false, a, false, bm, (short)0, acc[t], false, false);
        }
    }

    // D layout: vgpr v -> M = v + 8*half, N = l16 (+ tile offset).
    // Fast path: whole 16-row tile in bounds (all blocks except possibly the
    // last) -> unguarded stores with immediate offsets off one base pointer.
    float* hbase = h + (size_t)(row_base + 8 * half) * DIM + l16;
    if (row_base + 16 <= n_nodes) {
#pragma unroll
        for (int t = 0; t < 8; ++t) {
#pragma unroll
            for (int v = 0; v < 8; ++v) {
                hbase[(size_t)v * DIM + t * 16] = fast_tanh(acc[t][v]);
            }
        }
    } else {
#pragma unroll
        for (int t = 0; t < 8; ++t) {
#pragma unroll
            for (int v = 0; v < 8; ++v) {
                const int row = row_base + v + 8 * half;
                if (row < n_nodes)
                    h[(size_t)row * DIM + t * 16 + l16] = fast_tanh(acc[t][v]);
            }
        }
    }
}

// ---------------------------------------------------------------------------
// Zero the accumulator buffer (float4 grid-stride).
// ---------------------------------------------------------------------------
__global__ __launch_bounds__(256) void gcn_zero(float4* __restrict__ p, long n4)
{
    long i = (long)blockIdx.x * blockDim.x + threadIdx.x;
    long stride = (long)gridDim.x * blockDim.x;
    float4 z = make_float4(0.f, 0.f, 0.f, 0.f);
    for (; i < n4; i += stride) p[i] = z;
}

// ---------------------------------------------------------------------------
// Phase 2: COO SpMM scatter. One edge per wave32, one float4 per lane.
// agg[dst] += val * h[src] via hardware global_atomic_add_f32 (L2-resident:
// h + agg = 102 MB < 192 MB L2).
// ---------------------------------------------------------------------------
__global__ __launch_bounds__(256) void gcn_spmm_scatter(
    const float* __restrict__ h, const int* __restrict__ esrc,
    const int* __restrict__ edst, const float* __restrict__ eval,
    float* __restrict__ agg, int n_edges)
{
    const long gid  = (long)blockIdx.x * blockDim.x + threadIdx.x;
    const int  lane = (int)(gid & 31);
    const long e    = gid >> 5;
    if (e >= n_edges) return;

    const int   src = esrc[e];      // same address across the wave -> scalarized
    const int   dst = edst[e];
    const float val = eval[e];

    const float4 hv = ((const float4*)(h + (size_t)src * DIM))[lane];
    float* out = agg + (size_t)dst * DIM + lane * 4;
    unsafeAtomicAdd(out + 0, hv.x * val);
    unsafeAtomicAdd(out + 1, hv.y * val);
    unsafeAtomicAdd(out + 2, hv.z * val);
    unsafeAtomicAdd(out + 3, hv.w * val);
}

// ---------------------------------------------------------------------------
// Phase 3: out = tanh(agg) + b  (float4 vectorized; 32 float4 per row of 128)
// ---------------------------------------------------------------------------
__global__ __launch_bounds__(256) void gcn_finalize(
    const float* __restrict__ agg, const float* __restrict__ b,
    float* __restrict__ out, long n4)
{
    long i = (long)blockIdx.x * blockDim.x + threadIdx.x;
    if (i >= n4) return;
    float4 a  = ((const float4*)agg)[i];
    float4 bb = ((const float4*)b)[i & 31];   // 128 cols = 32 float4
    float4 r;
    r.x = fast_tanh(a.x) + bb.x;
    r.y = fast_tanh(a.y) + bb.y;
    r.z = fast_tanh(a.z) + bb.z;
    r.w = fast_tanh(a.w) + bb.w;
    ((float4*)out)[i] = r;
}

// ---------------------------------------------------------------------------
extern "C" void kernel_launch(void* const* d_in, const int* in_sizes, int n_in,
                              void* d_out, int out_size, void* d_ws, size_t ws_size,
                              hipStream_t stream)
{
    const float* x    = (const float*)d_in[0];
    const int*   esrc = (const int*)  d_in[1];
    const int*   edst = (const int*)  d_in[2];
    const float* eval = (const float*)d_in[3];
    const float* W    = (const float*)d_in[4];
    const float* b    = (const float*)d_in[5];

    const int n_nodes = in_sizes[0] / DIM;
    const int n_edges = in_sizes[1];

    // Workspace: h [N,D] then agg [N,D], both f32 (102.4 MB total).
    float* h   = (float*)d_ws;
    float* agg = h + (size_t)n_nodes * DIM;

    const long n4 = (long)n_nodes * (DIM / 4);

    // Zero agg (independent of GEMM; stream order guarantees it precedes SpMM).
    {
        long blocks = (n4 + 255) / 256;
        if (blocks > 4096) blocks = 4096;
        gcn_zero<<<(int)blocks, 256, 0, stream>>>((float4*)agg, n4);
    }

    // h = tanh(x @ W)
    {
        int blocks = (n_nodes + 127) / 128;
        gcn_gemm_tanh_wmma<<<blocks, 256, 0, stream>>>(x, W, h, n_nodes);
    }

    // agg[dst] += val * h[src]
    {
        long total = (long)n_edges * 32;
        long blocks = (total + 255) / 256;
        gcn_spmm_scatter<<<(int)blocks, 256, 0, stream>>>(h, esrc, edst, eval,
                                                          agg, n_edges);
    }

    // out = tanh(agg) + b
    {
        long blocks = (n4 + 255) / 256;
        gcn_finalize<<<(int)blocks, 256, 0, stream>>>(agg, b, (float*)d_out, n4);
    }
}